// RNNDecoder_4183298146479
// MI455X (gfx1250) — compile-verified
//
#include <hip/hip_runtime.h>
#include <math.h>

// Problem constants (match reference)
#define VOCAB 32000
#define EMB   512
#define HID   512
#define BB    32
#define SS    128
#define TT    64
#define KDIM  512
#define EPSC  1e-35f

typedef __attribute__((ext_vector_type(2))) float v2f;
typedef __attribute__((ext_vector_type(8))) float v8f;

// D = A(16x4 f32) * B(4x16 f32) + C(16x16 f32)  -- V_WMMA_F32_16X16X4_F32
static __device__ __forceinline__ v8f wmma_f32(v2f a, v2f b, v8f c) {
  return __builtin_amdgcn_wmma_f32_16x16x4_f32(false, a, false, b, (short)0, c,
                                               false, false);
}

// Load a 16x4 fragment from a row-major matrix where the fragment "rows" are
// rows row0..row0+15 and columns k..k+3 (stride = row stride in floats).
// A layout (ISA 7.12.2): lanes 0-15 -> {K=k,k+1}, lanes 16-31 -> {K=k+2,k+3}.
// The B fragment of W^T (B[k][n] = W[n][k]) has the identical access pattern,
// so this helper serves both sides.
static __device__ __forceinline__ v2f fragRM(const float* __restrict__ p,
                                             int row0, int k, int lane, int stride) {
  int r  = row0 + (lane & 15);
  int kk = k + ((lane >> 4) << 1);
  const float* q = p + (size_t)r * stride + kk;
  v2f f; f.x = q[0]; f.y = q[1];
  return f;
}

// ---------------------------------------------------------------------------
// K1: h0 = c0 = tanh(init_hidden @ bridge_W^T + bridge_b)   [32,512]x[512,512]
// 64 waves, one 16x16 tile each. Launch <<<8,256>>>.
// ---------------------------------------------------------------------------
__global__ void k_bridge(const float* __restrict__ ih, const float* __restrict__ bw,
                         const float* __restrict__ bb,
                         float* __restrict__ h0, float* __restrict__ c0) {
  int w = (blockIdx.x * blockDim.x + threadIdx.x) >> 5;
  int lane = threadIdx.x & 31;
  int mt = w >> 5, nt = w & 31;
  int row0 = mt * 16, col0 = nt * 16;
  v8f acc = {};
  for (int k = 0; k < KDIM; k += 4) {
    v2f a = fragRM(ih, row0, k, lane, KDIM);
    v2f b = fragRM(bw, col0, k, lane, KDIM);
    acc = wmma_f32(a, b, acc);
  }
  int col = col0 + (lane & 15);
  float bias = bb[col];
  int rb = row0 + ((lane >> 4) << 3);
#pragma unroll
  for (int r2 = 0; r2 < 8; ++r2) {
    float v = tanhf(acc[r2] + bias);
    h0[(rb + r2) * HID + col] = v;
    c0[(rb + r2) * HID + col] = v;
  }
}

// ---------------------------------------------------------------------------
// K2: u_feat[b,s,:] = tanh(enc_hs[s,b,:] @ copy_W^T + copy_b)
// M = B*S = 4096 rows (m = b*S + s), N = 512, K = 512. 8192 waves.
// Launch <<<1024,256>>>.
// ---------------------------------------------------------------------------
__global__ void k_ufeat(const float* __restrict__ enc_hs, const float* __restrict__ cw,
                        const float* __restrict__ cb, float* __restrict__ uf) {
  int w = (blockIdx.x * blockDim.x + threadIdx.x) >> 5;
  int lane = threadIdx.x & 31;
  int mt = w >> 5, nt = w & 31;                 // 256 M-tiles, 32 N-tiles
  int row0 = mt * 16, col0 = nt * 16;
  int m = row0 + (lane & 15);                   // m = b*S + s
  int b = m >> 7, s = m & (SS - 1);
  const float* arow = enc_hs + (size_t)(s * BB + b) * EMB;  // enc_hs[s,b,:]
  int koff = (lane >> 4) << 1;
  v8f acc = {};
  for (int k = 0; k < KDIM; k += 4) {
    v2f a; a.x = arow[k + koff]; a.y = arow[k + koff + 1];
    v2f bf = fragRM(cw, col0, k, lane, KDIM);
    acc = wmma_f32(a, bf, acc);
  }
  int col = col0 + (lane & 15);
  float bias = cb[col];
  int rb = row0 + ((lane >> 4) << 3);
#pragma unroll
  for (int r2 = 0; r2 < 8; ++r2)
    uf[(size_t)(rb + r2) * HID + col] = tanhf(acc[r2] + bias);
}

// ---------------------------------------------------------------------------
// K3: one LSTM step. Each wave owns a (b-tile, j-tile) and accumulates all
// four gate quadrants (i,f,g,o) so the elementwise LSTM update is done
// in-wave (8 WMMAs per K-step). 64 waves. Launch <<<8,256>>> per t.
// Reads hin/cin (distinct buffers), writes hout (= h_all[t]) and cout.
// ---------------------------------------------------------------------------
__global__ void k_lstm_step(const float* __restrict__ emb, const int* __restrict__ trg,
                            const float* __restrict__ wih, const float* __restrict__ whh,
                            const float* __restrict__ bih, const float* __restrict__ bhh,
                            const float* __restrict__ hin, const float* __restrict__ cin,
                            float* __restrict__ cout_, float* __restrict__ hout, int t) {
  int w = (blockIdx.x * blockDim.x + threadIdx.x) >> 5;
  int lane = threadIdx.x & 31;
  int mt = w >> 5, jt = w & 31;                 // 2 b-tiles, 32 j-tiles
  int row0 = mt * 16, col0 = jt * 16;
  int b = row0 + (lane & 15);
  int tok = (t == 0) ? 0 : trg[b * TT + (t - 1)];   // BOS = 0
  const float* xrow = emb + (size_t)tok * EMB;
  const float* hrow = hin + (size_t)b * HID;
  int koff = (lane >> 4) << 1;
  v8f acc[4] = {};
  for (int k = 0; k < KDIM; k += 4) {
    v2f ax; ax.x = xrow[k + koff]; ax.y = xrow[k + koff + 1];
    v2f ah; ah.x = hrow[k + koff]; ah.y = hrow[k + koff + 1];
#pragma unroll
    for (int q = 0; q < 4; ++q) {
      v2f bi = fragRM(wih, q * HID + col0, k, lane, KDIM);
      acc[q] = wmma_f32(ax, bi, acc[q]);
      v2f bh = fragRM(whh, q * HID + col0, k, lane, KDIM);
      acc[q] = wmma_f32(ah, bh, acc[q]);
    }
  }
  int col = col0 + (lane & 15);
  float bi0 = bih[0 * HID + col] + bhh[0 * HID + col];
  float bi1 = bih[1 * HID + col] + bhh[1 * HID + col];
  float bi2 = bih[2 * HID + col] + bhh[2 * HID + col];
  float bi3 = bih[3 * HID + col] + bhh[3 * HID + col];
  int rb = row0 + ((lane >> 4) << 3);
#pragma unroll
  for (int r2 = 0; r2 < 8; ++r2) {
    int bb_ = rb + r2;
    float gi = acc[0][r2] + bi0;
    float gf = acc[1][r2] + bi1;
    float gg = acc[2][r2] + bi2;
    float go = acc[3][r2] + bi3;
    float si = 1.f / (1.f + expf(-gi));
    float sf = 1.f / (1.f + expf(-gf));
    float so = 1.f / (1.f + expf(-go));
    float cold = cin[bb_ * HID + col];
    float cnew = sf * cold + si * tanhf(gg);
    cout_[bb_ * HID + col] = cnew;
    hout[bb_ * HID + col]  = so * tanhf(cnew);
  }
}

// ---------------------------------------------------------------------------
// K4: u[t,b,s] = u_feat[b,s,:] . h_all[t,b,:]  -- B batched GEMMs [64x128]xK512
// 32 b * 4 t-tiles * 8 s-tiles = 1024 waves. Launch <<<128,256>>>.
// ---------------------------------------------------------------------------
__global__ void k_uscore(const float* __restrict__ h_all, const float* __restrict__ uf,
                         float* __restrict__ u) {
  int w = (blockIdx.x * blockDim.x + threadIdx.x) >> 5;
  int lane = threadIdx.x & 31;
  int b = w >> 5;
  int rest = w & 31;
  int mt = rest >> 3, nt = rest & 7;
  int row0 = mt * 16, col0 = nt * 16;
  int tr = row0 + (lane & 15);
  int sc = col0 + (lane & 15);
  const float* arow = h_all + (size_t)(tr * BB + b) * HID;
  const float* brow = uf + (size_t)(b * SS + sc) * HID;
  int koff = (lane >> 4) << 1;
  v8f acc = {};
  for (int k = 0; k < KDIM; k += 4) {
    v2f a;  a.x  = arow[k + koff]; a.y  = arow[k + koff + 1];
    v2f bf; bf.x = brow[k + koff]; bf.y = brow[k + koff + 1];
    acc = wmma_f32(a, bf, acc);
  }
  int col = col0 + (lane & 15);
  int rb = row0 + ((lane >> 4) << 3);
#pragma unroll
  for (int r2 = 0; r2 < 8; ++r2)
    u[(size_t)((rb + r2) * BB + b) * SS + col] = acc[r2];
}

// ---------------------------------------------------------------------------
// K5: score_gen = h_all @ out_W^T + out_b  -> written straight into d_out.
// [2048 x 32000] x K=512. Block = 128 rows x 64 cols; the 64x512 out_W panel
// (128 KB) is staged once into LDS (row-padded to 516 floats: 16B-aligned rows
// for ds_store_b128 staging and conflict-free ds_load_b64 fragment reads),
// then 8 waves each run a 16x64 strip out of LDS. Cuts out_W L2 traffic 8x.
// Grid: 16 row-blocks * 500 col-groups = 8000 blocks. Launch
// <<<8000,256, 64*(KDIM+4)*4>>>.
// ---------------------------------------------------------------------------
#define LDSTR (KDIM + 4)   // 516: 516%64==4 -> fragment lanes hit distinct banks

__global__ void k_gen(const float* __restrict__ h_all, const float* __restrict__ ow,
                      const float* __restrict__ ob, float* __restrict__ out) {
  extern __shared__ float ldsB[];              // [64][LDSTR]
  int row_b = blockIdx.x / 500;
  int col_g = blockIdx.x % 500;
  int rb0  = row_b * 128;
  int col0 = col_g * 64;
  int tid = threadIdx.x;

  // Stage B panel: 64 vocab rows x 512 floats (uniform trip count, no
  // divergence ahead of the WMMAs).
  for (int idx = tid; idx < 64 * (KDIM / 4); idx += blockDim.x) {
    int c  = idx >> 7;            // local vocab row 0..63
    int kq = idx & 127;           // float4 index within row
    float4 v = ((const float4*)(ow + (size_t)(col0 + c) * KDIM))[kq];
    *(float4*)(ldsB + c * LDSTR + 4 * kq) = v;   // ds_store_b128 (16B aligned)
  }
  __syncthreads();

  int lane = tid & 31;
  int wv = tid >> 5;
  int row0 = rb0 + wv * 16;
  const float* arow = h_all + (size_t)(row0 + (lane & 15)) * HID;
  int koff = (lane >> 4) << 1;
  v8f acc[4] = {};
  for (int k = 0; k < KDIM; k += 4) {
    int kk = k + koff;
    v2f a; a.x = arow[kk]; a.y = arow[kk + 1];
#pragma unroll
    for (int i = 0; i < 4; ++i) {
      const float* bp = ldsB + (16 * i + (lane & 15)) * LDSTR + kk;
      v2f bf; bf.x = bp[0]; bf.y = bp[1];       // ds_load_b64, conflict-free
      acc[i] = wmma_f32(a, bf, acc[i]);
    }
  }
  int rb = row0 + ((lane >> 4) << 3);
#pragma unroll
  for (int i = 0; i < 4; ++i) {
    int col = col0 + 16 * i + (lane & 15);
    float bias = ob[col];
#pragma unroll
    for (int r2 = 0; r2 < 8; ++r2)
      out[(size_t)(rb + r2) * VOCAB + col] = acc[i][r2] + bias;
  }
}

// ---------------------------------------------------------------------------
// K6: finalize one (t,b) row: scatter-copy via dense 128KB LDS accumulator,
// joint softmax over [score_gen(32000) | u_copy(32000)], logp in-place.
// Launch <<<T*B, 256, (VOCAB+SS+256)*4>>>.
// ---------------------------------------------------------------------------
static __device__ __forceinline__ float blk_max(float v, float* scr) {
  int tid = threadIdx.x;
  scr[tid] = v; __syncthreads();
  for (int o = blockDim.x >> 1; o > 0; o >>= 1) {
    if (tid < o) scr[tid] = fmaxf(scr[tid], scr[tid + o]);
    __syncthreads();
  }
  float r = scr[0]; __syncthreads();
  return r;
}
static __device__ __forceinline__ float blk_sum(float v, float* scr) {
  int tid = threadIdx.x;
  scr[tid] = v; __syncthreads();
  for (int o = blockDim.x >> 1; o > 0; o >>= 1) {
    if (tid < o) scr[tid] += scr[tid + o];
    __syncthreads();
  }
  float r = scr[0]; __syncthreads();
  return r;
}

__global__ void k_finalize(const float* __restrict__ u, const int* __restrict__ enc_idxs,
                           float* __restrict__ out) {
  extern __shared__ float smem[];
  float* s_total = smem;               // VOCAB floats (dense scatter target)
  float* s_ue    = smem + VOCAB;       // SS floats
  float* s_red   = smem + VOCAB + SS;  // blockDim floats (reductions)
  int r = blockIdx.x;                  // r = t*B + b
  int b = r & (BB - 1);
  int tid = threadIdx.x;
  const float* urow = u + (size_t)r * SS;
  float* grow = out + (size_t)r * VOCAB;

  // copy-score softmax pieces
  float lmax = -INFINITY;
  for (int s = tid; s < SS; s += blockDim.x) lmax = fmaxf(lmax, urow[s]);
  float umax = blk_max(lmax, s_red);

  float lsum = 0.f;
  for (int s = tid; s < SS; s += blockDim.x) {
    float e = expf(urow[s] - umax);
    s_ue[s] = e;
    lsum += e;
  }
  float sum_ue = blk_sum(lsum, s_red);

  for (int v = tid; v < VOCAB; v += blockDim.x) s_total[v] = 0.f;
  __syncthreads();
  for (int s = tid; s < SS; s += blockDim.x)
    atomicAdd(&s_total[enc_idxs[b * SS + s]], s_ue[s]);   // ds_add_f32
  __syncthreads();

  // global max over concat(score_gen, u_copy); u_copy max = umax+log(maxtot+eps)
  float lgmax = -INFINITY, ltmax = 0.f;
  for (int v = tid; v < VOCAB; v += blockDim.x) {
    lgmax = fmaxf(lgmax, grow[v]);
    ltmax = fmaxf(ltmax, s_total[v]);
  }
  float gmax = blk_max(lgmax, s_red);
  float tmax = blk_max(ltmax, s_red);
  float M = fmaxf(gmax, umax + logf(tmax + EPSC));

  // denominator: sum exp(gen-M) + exp(umax-M)*(sum_ue + V*eps)
  float lz = 0.f;
  for (int v = tid; v < VOCAB; v += blockDim.x) lz += expf(grow[v] - M);
  float zg = blk_sum(lz, s_red);
  float eu = expf(umax - M);
  float logZ = logf(zg + eu * (sum_ue + (float)VOCAB * EPSC));

  // logp = log(exp(gen-M) + (total+eps)*exp(umax-M)) - logZ, in place
  for (int v = tid; v < VOCAB; v += blockDim.x) {
    float val = expf(grow[v] - M) + (s_total[v] + EPSC) * eu;
    grow[v] = logf(val) - logZ;
  }
}

// ---------------------------------------------------------------------------
extern "C" void kernel_launch(void* const* d_in, const int* in_sizes, int n_in,
                              void* d_out, int out_size, void* d_ws, size_t ws_size,
                              hipStream_t stream) {
  const int*   trg         = (const int*)d_in[0];
  const float* init_hidden = (const float*)d_in[1];
  const float* enc_hs      = (const float*)d_in[2];
  const int*   enc_idxs    = (const int*)d_in[3];
  const float* emb         = (const float*)d_in[4];
  const float* W_ih        = (const float*)d_in[5];
  const float* W_hh        = (const float*)d_in[6];
  const float* b_ih        = (const float*)d_in[7];
  const float* b_hh        = (const float*)d_in[8];
  const float* bridge_W    = (const float*)d_in[9];
  const float* bridge_b    = (const float*)d_in[10];
  const float* out_W       = (const float*)d_in[11];
  const float* out_b       = (const float*)d_in[12];
  const float* copy_W      = (const float*)d_in[13];
  const float* copy_b      = (const float*)d_in[14];
  float* out = (float*)d_out;

  // Workspace layout (~13.3 MB of fp32):
  float* ws     = (float*)d_ws;
  float* h0buf  = ws;                        // B*H
  float* cbuf0  = h0buf + BB * HID;          // B*H
  float* cbuf1  = cbuf0 + BB * HID;          // B*H (also holds c0)
  float* h_all  = cbuf1 + BB * HID;          // T*B*H
  float* u_feat = h_all + (size_t)TT * BB * HID;   // B*S*H
  float* u      = u_feat + (size_t)BB * SS * HID;  // T*B*S

  k_bridge<<<8, 256, 0, stream>>>(init_hidden, bridge_W, bridge_b, h0buf, cbuf1);
  k_ufeat<<<1024, 256, 0, stream>>>(enc_hs, copy_W, copy_b, u_feat);

  for (int t = 0; t < TT; ++t) {
    const float* hin = (t == 0) ? h0buf : h_all + (size_t)(t - 1) * BB * HID;
    const float* cin = (t == 0) ? cbuf1 : ((t & 1) ? cbuf0 : cbuf1);
    float* cout_     = (t & 1) ? cbuf1 : cbuf0;
    k_lstm_step<<<8, 256, 0, stream>>>(emb, trg, W_ih, W_hh, b_ih, b_hh,
                                       hin, cin, cout_,
                                       h_all + (size_t)t * BB * HID, t);
  }

  k_uscore<<<128, 256, 0, stream>>>(h_all, u_feat, u);
  k_gen<<<8000, 256, 64 * LDSTR * sizeof(float), stream>>>(h_all, out_W, out_b, out);
  k_finalize<<<TT * BB, 256, (VOCAB + SS + 256) * sizeof(float), stream>>>(u, enc_idxs, out);
}